// ResNet_1803886265717
// MI455X (gfx1250) — compile-verified
//
#include <hip/hip_runtime.h>
#include <hip/hip_bf16.h>

typedef __attribute__((ext_vector_type(2))) float v2f;
typedef __attribute__((ext_vector_type(8))) float v8f;

#define BN 32
#define CH 64
#define HW 65536
#define KSEG 196
#define KP 208          // 196 padded to 13*16 for WMMA row tiles
#define TOPN 98
#define NC 1000
#define NCP 1008        // 63*16

__device__ __forceinline__ v8f wmma4(v2f a, v2f b, v8f c) {
    // V_WMMA_F32_16X16X4_F32 : D[16x16] = A[16x4] * B[4x16] + C
    return __builtin_amdgcn_wmma_f32_16x16x4_f32(false, a, false, b, (short)0, c,
                                                 false, false);
}

// ---------------------------------------------------------------- zero scratch
__global__ void zero_kernel(float* __restrict__ p, int n) {
    for (int i = blockIdx.x * blockDim.x + threadIdx.x; i < n;
         i += gridDim.x * blockDim.x)
        p[i] = 0.0f;
}

// ------------------------------------------------- segment-sum pooling (x read)
// 16 blocks per image, 256 threads = 4 groups of 64 lanes (one lane per channel)
__global__ void pool_kernel(const float* __restrict__ x, const int* __restrict__ mask,
                            float* __restrict__ gsum, float* __restrict__ gcnt) {
    __shared__ float s_acc[KSEG * CH];   // 50176 B
    __shared__ float s_cnt[KSEG];
    const int tid = threadIdx.x;
    for (int i = tid; i < KSEG * CH; i += 256) s_acc[i] = 0.0f;
    for (int i = tid; i < KSEG; i += 256) s_cnt[i] = 0.0f;
    __syncthreads();
    const int SPLITS = 16;
    const int b = blockIdx.x / SPLITS;
    const int pix0 = (blockIdx.x % SPLITS) * (HW / SPLITS);
    const int grp = tid >> 6;          // 4 pixel groups
    const int c = tid & 63;            // channel lane
    const long xbase = (long)b * CH * HW;
    for (int p = grp; p < HW / SPLITS; p += 4) {
        const int pix = pix0 + p;
        const int seg = mask[b * HW + pix];
        // feats[b,pix,c] is a raw reinterpret: contiguous 64-float vectors
        const float val = x[xbase + (long)pix * CH + c];
        atomicAdd(&s_acc[seg * CH + c], val);
        if (c == 0) atomicAdd(&s_cnt[seg], 1.0f);
    }
    __syncthreads();
    float* gs = gsum + (long)b * KSEG * CH;
    for (int i = tid; i < KSEG * CH; i += 256) atomicAdd(&gs[i], s_acc[i]);
    float* gc = gcnt + b * KSEG;
    for (int i = tid; i < KSEG; i += 256) atomicAdd(&gc[i], s_cnt[i]);
}

// ------------------------------------------------------------ avg (row-padded)
__global__ void avg_kernel(const float* __restrict__ gsum, const float* __restrict__ gcnt,
                           float* __restrict__ avg) {
    const int n = BN * KP * CH;
    for (int i = blockIdx.x * blockDim.x + threadIdx.x; i < n;
         i += gridDim.x * blockDim.x) {
        const int b = i / (KP * CH);
        const int r = (i / CH) % KP;
        const int c = i % CH;
        avg[i] = (r < KSEG)
                   ? gsum[((long)b * KSEG + r) * CH + c] / fmaxf(gcnt[b * KSEG + r], 1.0f)
                   : 0.0f;
    }
}

// ---------------------------------------------------- q/k/v projections (WMMA)
// job = (b, row-tile, matrix, col-tile); one 16x16 tile per wave
__global__ void qkv_kernel(const float* __restrict__ avg,
                           const float* __restrict__ Wq, const float* __restrict__ bq,
                           const float* __restrict__ Wk, const float* __restrict__ bk,
                           const float* __restrict__ Wv, const float* __restrict__ bv,
                           float* __restrict__ q, float* __restrict__ k,
                           float* __restrict__ v) {
    const int wave = threadIdx.x >> 5;
    const int lane = threadIdx.x & 31;
    const int half = lane >> 4, mn = lane & 15;
    const int NJOBS = BN * 13 * 12;
    const int job = blockIdx.x * 8 + wave;
    if (job >= NJOBS) return;                 // wave-uniform
    const int b = job / (13 * 12);
    const int rem = job % (13 * 12);
    const int mt = rem / 12;
    const int sub = rem % 12;
    const int mat = sub >> 2;
    const int nt = sub & 3;
    const float* Wm = (mat == 0) ? Wq : (mat == 1) ? Wk : Wv;
    const float* bm = (mat == 0) ? bq : (mat == 1) ? bk : bv;
    float* dst = (mat == 0) ? q : (mat == 1) ? k : v;
    const float* A = avg + (long)b * KP * CH;
    const int r0 = mt * 16, c0 = nt * 16;
    v8f acc = {};
    for (int k0 = 0; k0 < CH; k0 += 4) {
        const int kk = k0 + half * 2;
        v2f a, bb;
        a.x = A[(r0 + mn) * CH + kk];
        a.y = A[(r0 + mn) * CH + kk + 1];
        bb.x = Wm[kk * CH + c0 + mn];          // W stored [in,out]
        bb.y = Wm[(kk + 1) * CH + c0 + mn];
        acc = wmma4(a, bb, acc);
    }
    const float bias = bm[c0 + mn];
    float* D = dst + (long)b * KP * CH;
    for (int r = 0; r < 8; ++r) {
        const int row = r0 + half * 8 + r;
        D[row * CH + c0 + mn] = (row < KSEG) ? (acc[r] + bias) : 0.0f;  // pad rows = 0
    }
}

// -------------------------- scores + softmax + P@V + residual + LayerNorm (WMMA)
// one block per (batch, query row-tile)
__global__ void attn_kernel(const float* __restrict__ qbuf, const float* __restrict__ kbuf,
                            const float* __restrict__ vbuf, const float* __restrict__ avg,
                            const float* __restrict__ lnw, const float* __restrict__ lnb,
                            float* __restrict__ hn, float* __restrict__ attv) {
    __shared__ float s_sc[16 * KP];   // scores/probs for 16 query rows
    __shared__ float s_h[16 * CH];    // attention output tile
    const int b = blockIdx.x / 13;
    const int rt = blockIdx.x % 13;
    const int r0 = rt * 16;
    const int tid = threadIdx.x;
    const int wave = tid >> 5, lane = tid & 31;
    const int half = lane >> 4, mn = lane & 15;
    const float* qb = qbuf + (long)b * KP * CH;
    const float* kb = kbuf + (long)b * KP * CH;
    const float* vb = vbuf + (long)b * KP * CH;

    // scores[16][208] = Q_tile @ K^T / sqrt(64)
    for (int ct = wave; ct < 13; ct += 8) {
        const int c0 = ct * 16;
        v8f acc = {};
        for (int k0 = 0; k0 < CH; k0 += 4) {
            const int kk = k0 + half * 2;
            v2f a, bb;
            a.x = qb[(r0 + mn) * CH + kk];
            a.y = qb[(r0 + mn) * CH + kk + 1];
            bb.x = kb[(c0 + mn) * CH + kk];     // K transposed on the fly
            bb.y = kb[(c0 + mn) * CH + kk + 1];
            acc = wmma4(a, bb, acc);
        }
        for (int r = 0; r < 8; ++r)
            s_sc[(half * 8 + r) * KP + c0 + mn] = acc[r] * 0.125f;  // /sqrt(64)
    }
    __syncthreads();
    // row softmax (mask pad columns)
    if (tid < 16) {
        float* row = &s_sc[tid * KP];
        for (int j = KSEG; j < KP; ++j) row[j] = -1e30f;
        float mx = -1e30f;
        for (int j = 0; j < KSEG; ++j) mx = fmaxf(mx, row[j]);
        float sum = 0.0f;
        for (int j = 0; j < KP; ++j) { float e = __expf(row[j] - mx); row[j] = e; sum += e; }
        const float inv = 1.0f / sum;
        for (int j = 0; j < KP; ++j) row[j] *= inv;
    }
    __syncthreads();
    // h_tile[16][64] = P @ V
    if (wave < 4) {                            // wave-uniform
        const int c0 = wave * 16;
        v8f acc = {};
        for (int k0 = 0; k0 < KP; k0 += 4) {
            const int kk = k0 + half * 2;
            v2f a, bb;
            a.x = s_sc[mn * KP + kk];
            a.y = s_sc[mn * KP + kk + 1];
            bb.x = vb[kk * CH + c0 + mn];
            bb.y = vb[(kk + 1) * CH + c0 + mn];
            acc = wmma4(a, bb, acc);
        }
        for (int r = 0; r < 8; ++r)
            s_h[(half * 8 + r) * CH + c0 + mn] = acc[r];
    }
    __syncthreads();
    // residual + TF LayerNorm (eps inside sqrt) + row sum
    if (tid < 16) {
        const int grow = r0 + tid;
        if (grow < KSEG) {
            const float* avr = avg + (long)b * KP * CH + grow * CH;
            float u = 0.0f;
            for (int c = 0; c < CH; ++c) {
                const float hv = s_h[tid * CH + c] + avr[c];
                s_h[tid * CH + c] = hv;
                u += hv;
            }
            u *= (1.0f / CH);
            float s2 = 0.0f;
            for (int c = 0; c < CH; ++c) { const float d = s_h[tid * CH + c] - u; s2 += d * d; }
            s2 *= (1.0f / CH);
            const float rstd = rsqrtf(s2 + 1e-12f);
            float asum = 0.0f;
            float* out = hn + ((long)b * KSEG + grow) * CH;
            for (int c = 0; c < CH; ++c) {
                const float val = lnw[c] * ((s_h[tid * CH + c] - u) * rstd) + lnb[c];
                out[c] = val;
                asum += val;
            }
            attv[b * KSEG + grow] = asum;
        }
    }
}

// ------------------------------------------- top-k (rank count) + sigmoid + gather
__global__ void topk_kernel(const float* __restrict__ attv, const float* __restrict__ hn,
                            float* __restrict__ o_top, float* __restrict__ o_idx,
                            float* __restrict__ o_sp) {
    __shared__ float s_att[KSEG];
    __shared__ int s_sel[TOPN];
    const int b = blockIdx.x;
    const int tid = threadIdx.x;
    if (tid < KSEG) s_att[tid] = attv[b * KSEG + tid];
    __syncthreads();
    if (tid < KSEG) {
        const float ai = s_att[tid];
        int rank = 0;
        for (int j = 0; j < KSEG; ++j) {
            const float aj = s_att[j];
            rank += (aj > ai) || (aj == ai && j < tid);   // jax top_k tie-break
        }
        if (rank < TOPN) s_sel[rank] = tid;
        o_sp[b * KSEG + tid] = 1.0f / (1.0f + __expf(-ai));
    }
    __syncthreads();
    if (tid < TOPN) o_idx[b * TOPN + tid] = (float)s_sel[tid];
    for (int e = tid; e < TOPN * CH; e += blockDim.x) {
        const int t = e >> 6, c = e & 63;
        o_top[((long)b * TOPN + t) * CH + c] = hn[((long)b * KSEG + s_sel[t]) * CH + c];
    }
}

// ---------------------------------------------------- classifier GEMM (WMMA)
// [3136,64] @ [64,1000] ; 196 x 63 tiles, one tile per wave
__global__ void cls_kernel(const float* __restrict__ topfeat, const float* __restrict__ Wl,
                           const float* __restrict__ bl, float* __restrict__ out) {
    const int wave = threadIdx.x >> 5;
    const int lane = threadIdx.x & 31;
    const int half = lane >> 4, mn = lane & 15;
    const int MT = (BN * TOPN) / 16;   // 196
    const int NT = NCP / 16;           // 63
    const int job = blockIdx.x * 8 + wave;
    if (job >= MT * NT) return;        // wave-uniform
    const int mt = job / NT, nt = job % NT;
    const int r0 = mt * 16, c0 = nt * 16;
    const int col = c0 + mn;
    v8f acc = {};
    for (int k0 = 0; k0 < CH; k0 += 4) {
        const int kk = k0 + half * 2;
        v2f a, bb;
        a.x = topfeat[(r0 + mn) * CH + kk];
        a.y = topfeat[(r0 + mn) * CH + kk + 1];
        bb.x = (col < NC) ? Wl[kk * NC + col] : 0.0f;
        bb.y = (col < NC) ? Wl[(kk + 1) * NC + col] : 0.0f;
        acc = wmma4(a, bb, acc);
    }
    if (col < NC) {
        const float bias = bl[col];
        for (int r = 0; r < 8; ++r) {
            const int row = r0 + half * 8 + r;
            out[(long)row * NC + col] = acc[r] + bias;
        }
    }
}

extern "C" void kernel_launch(void* const* d_in, const int* in_sizes, int n_in,
                              void* d_out, int out_size, void* d_ws, size_t ws_size,
                              hipStream_t stream) {
    (void)in_sizes; (void)n_in; (void)out_size; (void)ws_size;
    const float* x   = (const float*)d_in[0];
    const int*   msk = (const int*)d_in[1];
    const float* Wq  = (const float*)d_in[2];  const float* bq = (const float*)d_in[3];
    const float* Wk  = (const float*)d_in[4];  const float* bk = (const float*)d_in[5];
    const float* Wv  = (const float*)d_in[6];  const float* bv = (const float*)d_in[7];
    const float* lnw = (const float*)d_in[8];  const float* lnb = (const float*)d_in[9];
    const float* Wl  = (const float*)d_in[10]; const float* bl = (const float*)d_in[11];

    float* out = (float*)d_out;
    float* ws  = (float*)d_ws;
    // workspace layout (floats)
    float* gsum = ws;                               // B*K*C   = 401408
    float* gcnt = gsum + BN * KSEG * CH;            // B*K     =   6272
    float* avg  = gcnt + BN * KSEG;                 // B*KP*C  = 425984
    float* q    = avg + BN * KP * CH;
    float* k    = q   + BN * KP * CH;
    float* v    = k   + BN * KP * CH;
    float* hn   = v   + BN * KP * CH;               // B*K*C
    float* attv = hn  + BN * KSEG * CH;             // B*K

    // d_out layout: local_cls | top_feat | idx | sp_attention
    float* o_cls = out;
    float* o_top = o_cls + (size_t)BN * TOPN * NC;
    float* o_idx = o_top + (size_t)BN * TOPN * CH;
    float* o_sp  = o_idx + (size_t)BN * TOPN;

    zero_kernel<<<512, 256, 0, stream>>>(gsum, BN * KSEG * CH + BN * KSEG);
    pool_kernel<<<BN * 16, 256, 0, stream>>>(x, msk, gsum, gcnt);
    avg_kernel<<<512, 256, 0, stream>>>(gsum, gcnt, avg);
    qkv_kernel<<<(BN * 13 * 12 + 7) / 8, 256, 0, stream>>>(avg, Wq, bq, Wk, bk, Wv, bv,
                                                           q, k, v);
    attn_kernel<<<BN * 13, 256, 0, stream>>>(q, k, v, avg, lnw, lnb, hn, attv);
    topk_kernel<<<BN, 256, 0, stream>>>(attv, hn, o_top, o_idx, o_sp);
    cls_kernel<<<(196 * 63 + 7) / 8, 256, 0, stream>>>(o_top, Wl, bl, o_cls);
}